// ArcLengthLoss_48876727829132
// MI455X (gfx1250) — compile-verified
//
#include <hip/hip_runtime.h>
#include <math.h>

// Problem constants (from reference: K=18, BATCH=32)
#define KSEC   153                 // NUM_SECTIONS = 18*17/2
#define KPAD   160                 // K padded to multiple of 4 for 16x16x4 WMMA
#define SSTRIDE 161                // LDS row stride (coprime with 64 banks)
#define NBATCH 32
#define NCOMBO 261972              // sum_{m=3..18} C(18,m)
#define LROW   17                  // pad_idx row length
#define NA     136                 // len(a1) == len(a2)
#define WAVES  4
#define TPB    (WAVES * 32)        // wave32: 4 waves / block
#define COMBOS_PER_BLOCK (WAVES * 16)

typedef __attribute__((ext_vector_type(2))) float v2f;
typedef __attribute__((ext_vector_type(8))) float v8f;

// b2 = s(32 x 153) * C^T(153 x NC) via fp32 WMMA, fused with |b1 - b2| partial sums.
// Each wave handles a 16-combo N-tile; M=32 split into two 16x16 tiles.
__global__ __launch_bounds__(TPB)
void arc_b2_kernel(const float* __restrict__ sflat,      // (32,153) fp32
                   const int*   __restrict__ direct,     // (NC,)
                   const int*   __restrict__ pad_idx,    // (NC,17)
                   const int*   __restrict__ seg_len,    // (NC,)
                   float*       __restrict__ partials) { // (gridDim.x,)
  __shared__ float s_lds[NBATCH * SSTRIDE];              // A operand, padded stride
  __shared__ float b_lds[WAVES * 16 * SSTRIDE];          // per-wave 0/1 B tiles
  __shared__ float red[TPB];

  const int tid   = threadIdx.x;
  const int wv    = tid >> 5;
  const int lane  = tid & 31;
  const int cbase = blockIdx.x * COMBOS_PER_BLOCK + wv * 16;

  // Stage s into LDS: [batch][k], zero-padded K 153..159
  for (int i = tid; i < NBATCH * KPAD; i += TPB) {
    const int b = i / KPAD, k = i % KPAD;
    s_lds[b * SSTRIDE + k] = (k < KSEC) ? sflat[b * KSEC + k] : 0.0f;
  }
  // Zero this wave's B tile
  float* bt = &b_lds[wv * 16 * SSTRIDE];
  for (int i = lane; i < 16 * KPAD; i += 32) {
    const int n = i / KPAD, k = i % KPAD;
    bt[n * SSTRIDE + k] = 0.0f;
  }
  __syncthreads();

  // Scatter the 0/1 selection rows: bt[n][k] = 1 iff section k lies on combo (cbase+n)'s path.
  // This is the HBM-bound stream (pad_idx, ~17.8 MB total across the grid).
  if (lane < 16) {
    const int c = cbase + lane;
    if (c < NCOMBO) {
      __builtin_prefetch(&direct[c], 0, 0);     // global_prefetch_b8 for the epilogue gather
      const int  sl   = seg_len[c];
      const int* prow = &pad_idx[c * LROW];
      for (int j = 0; j < sl; ++j)
        bt[lane * SSTRIDE + prow[j]] = 1.0f;
    }
  }
  __syncthreads();

  // fp32 WMMA: D(16x16) += A(16x4) x B(4x16), 40 K-steps.
  // A layout (ISA 16x4 f32): lanes 0-15 -> M=lane, K={k0,k0+1}; lanes 16-31 -> M=lane-16, K={k0+2,k0+3}.
  // B layout mirrors it over (K,N) with N=lane%16.
  v8f c0 = {}; // batches 0..15
  v8f c1 = {}; // batches 16..31
  const int kk  = (lane < 16) ? 0 : 2;
  const int row = lane & 15;
  for (int k0 = 0; k0 < KPAD; k0 += 4) {
    v2f a0, a1v, bb;
    a0.x  = s_lds[row * SSTRIDE + k0 + kk];
    a0.y  = s_lds[row * SSTRIDE + k0 + kk + 1];
    a1v.x = s_lds[(row + 16) * SSTRIDE + k0 + kk];
    a1v.y = s_lds[(row + 16) * SSTRIDE + k0 + kk + 1];
    bb.x  = bt[row * SSTRIDE + k0 + kk];
    bb.y  = bt[row * SSTRIDE + k0 + kk + 1];
    c0 = __builtin_amdgcn_wmma_f32_16x16x4_f32(false, a0,  false, bb, (short)0, c0, false, false);
    c1 = __builtin_amdgcn_wmma_f32_16x16x4_f32(false, a1v, false, bb, (short)0, c1, false, false);
  }

  // Epilogue: C/D layout -> VGPR r holds (M=r | r+8, N=lane%16). Fuse |b1 - b2|.
  float part = 0.0f;
  {
    const int c = cbase + (lane & 15);
    if (c < NCOMBO) {
      const int dc = direct[c];
      #pragma unroll
      for (int r = 0; r < 8; ++r) {
        const int m0 = (lane < 16) ? r : (r + 8);
        part += fabsf(s_lds[m0 * SSTRIDE + dc]        - c0[r]);
        part += fabsf(s_lds[(m0 + 16) * SSTRIDE + dc] - c1[r]);
      }
    }
  }

  // Deterministic block reduction (no atomics)
  red[tid] = part;
  __syncthreads();
  for (int off = TPB / 2; off > 0; off >>= 1) {
    if (tid < off) red[tid] += red[tid + off];
    __syncthreads();
  }
  if (tid == 0) partials[blockIdx.x] = red[0];
}

// Sum per-block partials, add the tiny a_term, scale, write scalar.
__global__ __launch_bounds__(256)
void arc_finalize_kernel(const float* __restrict__ sflat,
                         const int*   __restrict__ a1,
                         const int*   __restrict__ a2,
                         const float* __restrict__ partials, int nparts,
                         float*       __restrict__ out) {
  __shared__ float red[256];
  const int tid = threadIdx.x;

  float acc = 0.0f;
  for (int i = tid; i < nparts; i += 256) acc += partials[i];
  red[tid] = acc;
  __syncthreads();
  for (int off = 128; off > 0; off >>= 1) {
    if (tid < off) red[tid] += red[tid + off];
    __syncthreads();
  }
  const float bsum = red[0];
  __syncthreads();

  float aval = 0.0f;
  if (tid < NBATCH) {
    float s = 0.0f;
    for (int i = 0; i < NA; ++i)
      s += sflat[tid * KSEC + a1[i]] - sflat[tid * KSEC + a2[i]];
    aval = expf(s / (float)NA);
  }
  red[tid] = aval;
  __syncthreads();
  for (int off = 128; off > 0; off >>= 1) {
    if (tid < off) red[tid] += red[tid + off];
    __syncthreads();
  }
  if (tid == 0)
    out[0] = red[0] / (float)NBATCH + bsum / ((float)NBATCH * (float)NCOMBO);
}

extern "C" void kernel_launch(void* const* d_in, const int* in_sizes, int n_in,
                              void* d_out, int out_size, void* d_ws, size_t ws_size,
                              hipStream_t stream) {
  (void)in_sizes; (void)n_in; (void)out_size; (void)ws_size;
  const float* s       = (const float*)d_in[0]; // output (32,153,1)
  const int*   a1      = (const int*)  d_in[1];
  const int*   a2      = (const int*)  d_in[2];
  const int*   direct  = (const int*)  d_in[3];
  const int*   pad_idx = (const int*)  d_in[4];
  const int*   seg_len = (const int*)  d_in[5];
  float* partials = (float*)d_ws;               // gridDim floats of scratch

  const int nblocks = (NCOMBO + COMBOS_PER_BLOCK - 1) / COMBOS_PER_BLOCK; // 4094
  arc_b2_kernel<<<dim3(nblocks), dim3(TPB), 0, stream>>>(s, direct, pad_idx, seg_len, partials);
  arc_finalize_kernel<<<dim3(1), dim3(256), 0, stream>>>(s, a1, a2, partials, nblocks, (float*)d_out);
}